// CTCLayer_66279935312436
// MI455X (gfx1250) — compile-verified
//
#include <hip/hip_runtime.h>
#include <hip/hip_bf16.h>

// CTC forward loss (keras ctc_batch_cost semantics) for fixed shapes:
//   B=512, T=512, C=100 (blank = C-1 = 99), L=128, S = 2L+1 = 257.
// One wave32 per batch element. Alpha lattice lives entirely in registers
// (9 states per lane), recursion runs in the log2 domain so logsumexp uses
// raw v_exp_f32 / v_log_f32, and the 3-way logsumexp uses max3/med3/min3
// (the max term contributes exp2(0)=1, saving one v_exp per state).
// Per-timestep class row (400B) is double-buffered in LDS via CDNA5 async
// global->LDS copies with s_wait_asynccnt 1 pipelining.

#define CTC_T     512
#define CTC_C     100
#define CTC_L     128
#define CTC_S     257          // 2L+1
#define CTC_BLANK 99           // C-1
#define JPL       9            // states per lane: 32*9 = 288 >= 257
#define NEGV      (-1e30f)
#define EPSV      (1e-7f)
#define LN2F      (0.69314718055994530942f)
#define ROW_BYTES 512u         // LDS slot stride (128 floats; 100 used)

// ---- CDNA5 async global->LDS (ASYNCcnt-tracked) -------------------------
__device__ __forceinline__ unsigned lds_off_u32(const void* p) {
    // LDS aperture keeps the wave-relative LDS byte offset in addr[31:0].
    return (unsigned)(unsigned long long)p;
}

__device__ __forceinline__ void async_copy_b128(unsigned lds_byte_off,
                                                const void* gaddr) {
    asm volatile("global_load_async_to_lds_b128 %0, %1, off"
                 :: "v"(lds_byte_off),
                    "v"((unsigned long long)gaddr)
                 : "memory");
}

__device__ __forceinline__ void wait_async0() {
    asm volatile("s_wait_asynccnt 0" ::: "memory");
}
// Async loads complete in order: count<=1 => the older prefetch has landed.
__device__ __forceinline__ void wait_async1() {
    asm volatile("s_wait_asynccnt 1" ::: "memory");
}

// raw hardware transcendentals (base-2)
__device__ __forceinline__ float hw_exp2(float x) { return __builtin_amdgcn_exp2f(x); }
__device__ __forceinline__ float hw_log2(float x) { return __builtin_amdgcn_logf(x); }

// ------------------------------------------------------------------------
__global__ __launch_bounds__(32)
void ctc_forward_kernel(const int* __restrict__ y_true,    // (B, L) int32
                        const float* __restrict__ y_pred,  // (B, T, C) f32
                        float* __restrict__ out)           // (B, 1)   f32
{
    __shared__ float rowbuf[2][128];   // two 512B slots, 400B used each

    const int b    = blockIdx.x;
    const int lane = threadIdx.x;      // wave32

    const int*   lab  = y_true + (size_t)b * CTC_L;
    const float* base = y_pred + (size_t)b * CTC_T * CTC_C;

    const unsigned lds_base = lds_off_u32(&rowbuf[0][0]);

    // Copy lanes: 25 x 16B = 400B. Lanes 25..31 duplicate lane 24 (same
    // src, same dst, same data -> benign) to avoid per-iteration EXEC branches.
    const int      cl       = (lane < 25) ? lane : 24;
    const unsigned lds_lane = (unsigned)cl * 16u;
    const float*   glane    = base + cl * 4;

    // ---- kick off async prefetch of row t=0 while we build ext[] --------
    async_copy_b128(lds_base + lds_lane, glane);

    // ---- per-lane extended-label classes + skip adders (registers) ------
    int   cls[JPL];
    float skipadd[JPL];                // 0.0f if skip allowed, NEGV if not
    #pragma unroll
    for (int j = 0; j < JPL; ++j) {
        int s = lane * JPL + j;
        int c = CTC_BLANK;
        bool skip = false;
        if (s < CTC_S) {
            if (s & 1) {
                int li = (s - 1) >> 1;
                c = lab[li];
                if (s >= 3) skip = (c != lab[li - 1]);
            }
        }
        cls[j]     = c;
        skipadd[j] = skip ? 0.0f : NEGV;
    }

    // ---- alpha_0 (log2 domain) ------------------------------------------
    wait_async0();                      // row 0 resident in LDS
    float ap[JPL];
    #pragma unroll
    for (int j = 0; j < JPL; ++j) {
        int s = lane * JPL + j;
        float a = NEGV;
        if (s < 2) a = hw_log2(rowbuf[0][cls[j]] + EPSV);
        ap[j] = a;
    }

    // prefetch row t=1 into slot 1
    async_copy_b128(lds_base + ROW_BYTES + lds_lane, glane + CTC_C);

    // ---- forward recursion over T-1 steps -------------------------------
    #pragma unroll 2
    for (int t = 1; t < CTC_T; ++t) {
        if (t + 1 < CTC_T) {
            // issue prefetch of row t+1 first, then wait for row t only
            unsigned slot = (unsigned)((t + 1) & 1) * ROW_BYTES;
            async_copy_b128(lds_base + slot + lds_lane,
                            glane + (size_t)(t + 1) * CTC_C);
            wait_async1();
        } else {
            wait_async0();
        }

        // cross-lane boundary alphas: states 9l-1 and 9l-2 from lane l-1
        float up8 = __shfl_up(ap[8], 1);   // alpha[9l-1]
        float up7 = __shfl_up(ap[7], 1);   // alpha[9l-2]
        if (lane == 0) { up8 = NEGV; up7 = NEGV; }

        const float* row = (t & 1) ? &rowbuf[1][0] : &rowbuf[0][0];

        float an[JPL];
        #pragma unroll
        for (int j = 0; j < JPL; ++j) {
            float a1 = ap[j];
            float a2 = (j == 0) ? up8 : ap[j - 1];
            float a3 = ((j == 0) ? up7 : ((j == 1) ? up8 : ap[j - 2]))
                       + skipadd[j];                   // masked -> <= -1e30

            // 3-way logsumexp: the max term is exp2(0)=1 -> only 2 exps.
            float m   = fmaxf(a1, fmaxf(a2, a3));              // v_max3_f32
            float md  = __builtin_amdgcn_fmed3f(a1, a2, a3);   // v_med3_f32
            float mn  = fminf(a1, fminf(a2, a3));              // v_min3_f32
            float sum = 1.0f + hw_exp2(md - m) + hw_exp2(mn - m);

            float lp2  = hw_log2(row[cls[j]] + EPSV);  // ds_load gather
            float head = m + lp2;                      // overlaps v_log below
            an[j] = head + hw_log2(sum);
        }
        #pragma unroll
        for (int j = 0; j < JPL; ++j) ap[j] = an[j];
    }

    // ---- -logaddexp(alpha[S-1], alpha[S-2]); states 256,255 on lane 28 --
    // alpha is log2-domain: nats = log2_value * ln(2).
    if (lane == 28) {
        float aL = ap[3];               // s = 255 (last label)
        float aB = ap[4];               // s = 256 (final blank)
        float m  = fmaxf(aL, aB);
        float ll2 = m + hw_log2(hw_exp2(aL - m) + hw_exp2(aB - m));
        out[b] = -LN2F * ll2;
    }
}

// ------------------------------------------------------------------------
extern "C" void kernel_launch(void* const* d_in, const int* in_sizes, int n_in,
                              void* d_out, int out_size, void* d_ws, size_t ws_size,
                              hipStream_t stream) {
    (void)in_sizes; (void)n_in; (void)d_ws; (void)ws_size;
    const int*   y_true = (const int*)d_in[0];     // (B, L) int32
    const float* y_pred = (const float*)d_in[1];   // (B, T, C) float32
    float*       out    = (float*)d_out;           // (B, 1) float32

    const int B = out_size;                        // 512
    ctc_forward_kernel<<<B, 32, 0, stream>>>(y_true, y_pred, out);
}